// GNNPrototype_15668040696097
// MI455X (gfx1250) — compile-verified
//
#include <hip/hip_runtime.h>
#include <stdint.h>

// ---- problem constants (match reference) ----
#define N_NODES 100000
#define N_EDGES 1600000
#define F_IN    256
#define F_HID   128
#define F_OUT   16
#define MTILES  (N_NODES / 16)   // 6250 row tiles of 16 nodes

typedef float v2f __attribute__((ext_vector_type(2)));
typedef float v8f __attribute__((ext_vector_type(8)));

// ---------------------------------------------------------------------------
// degree / normalization
// ---------------------------------------------------------------------------
__global__ void k_init_deg(float* __restrict__ deg) {
    int n = blockIdx.x * 256 + threadIdx.x;
    if (n < N_NODES) deg[n] = 1.0f;   // self-loop contributes 1 to every node
}

__global__ void k_count_deg(const long long* __restrict__ dst, float* __restrict__ deg) {
    int e = blockIdx.x * 256 + threadIdx.x;
    if (e < N_EDGES) {
        __hip_atomic_fetch_add(&deg[(int)dst[e]], 1.0f,
                               __ATOMIC_RELAXED, __HIP_MEMORY_SCOPE_AGENT);
    }
}

__global__ void k_dinv(float* __restrict__ deg) {
    int n = blockIdx.x * 256 + threadIdx.x;
    if (n < N_NODES) {
        float d = deg[n];
        deg[n] = (d > 0.0f) ? rsqrtf(d) : 0.0f;   // deg >= 1 here, kept for fidelity
    }
}

__global__ void k_zero4(float4* __restrict__ p, int n4) {
    int i = blockIdx.x * 256 + threadIdx.x;
    if (i < n4) p[i] = make_float4(0.f, 0.f, 0.f, 0.f);
}

// ---------------------------------------------------------------------------
// GEMM1: H1 = X @ W1   (100000x256 @ 256x128), fp32 WMMA 16x16x4
// block = 256 threads = 8 waves; one 16-node row tile per block,
// wave w computes output columns [16w, 16w+16).
// X tile (16x256) staged in LDS, padded stride 257 -> conflict-free A reads.
// ---------------------------------------------------------------------------
__global__ __launch_bounds__(256) void k_gemm1(const float* __restrict__ X,
                                               const float* __restrict__ W1,
                                               float* __restrict__ H1) {
    __shared__ float ldsX[16 * 257];
    const int t  = threadIdx.x;
    const int Mb = blockIdx.x * 16;

    // cooperative coalesced fill of the 16x256 X tile
#pragma unroll
    for (int i = 0; i < 16; ++i) {
        int idx = t + 256 * i;
        int row = idx >> 8;
        int col = idx & 255;
        ldsX[row * 257 + col] = X[(size_t)(Mb + row) * F_IN + col];
    }
    __syncthreads();

    const int lane = t & 31;
    const int wave = t >> 5;
    const int r    = lane & 15;            // A: row (M); B: column (N)
    const int koff = (lane >> 4) << 1;     // lanes 16-31 hold K=2,3
    const int Nb   = wave * 16;

    v8f acc = {};
#pragma unroll 8
    for (int k0 = 0; k0 < F_IN; k0 += 4) {
        v2f a, b;
        a.x = ldsX[r * 257 + k0 + koff];
        a.y = ldsX[r * 257 + k0 + koff + 1];
        b.x = W1[(size_t)(k0 + koff)     * F_HID + Nb + r];
        b.y = W1[(size_t)(k0 + koff + 1) * F_HID + Nb + r];
        acc = __builtin_amdgcn_wmma_f32_16x16x4_f32(
                  false, a, false, b, (short)0, acc, false, false);
    }

    // C/D layout: vgpr j -> M = j + 8*(lane>=16), N = lane&15
    const int col   = Nb + r;
    const int rbase = Mb + ((lane >> 4) << 3);
#pragma unroll
    for (int j = 0; j < 8; ++j)
        H1[(size_t)(rbase + j) * F_HID + col] = acc[j];
}

// ---------------------------------------------------------------------------
// scatter layer 1: agg1[dst] += H1[src] * dinv[src]*dinv[dst]
// one thread per (edge, channel); edges E..E+N-1 are the self-loops.
// hardware f32 atomics resolve in L2 (agg1 = 51 MB << 192 MB L2).
// ---------------------------------------------------------------------------
__global__ void k_scatter1(const long long* __restrict__ src,
                           const long long* __restrict__ dst,
                           const float* __restrict__ dinv,
                           const float* __restrict__ H1,
                           float* __restrict__ agg1) {
    long long gid = (long long)blockIdx.x * 256 + threadIdx.x;
    int e = (int)(gid >> 7);
    int c = (int)(gid & 127);
    if (e >= N_EDGES + N_NODES) return;
    int s, d;
    if (e < N_EDGES) { s = (int)src[e]; d = (int)dst[e]; }
    else             { s = d = e - N_EDGES; }
    float w = dinv[s] * dinv[d];
    float v = H1[(size_t)s * F_HID + c] * w;
    __hip_atomic_fetch_add(&agg1[(size_t)d * F_HID + c], v,
                           __ATOMIC_RELAXED, __HIP_MEMORY_SCOPE_AGENT);
}

// ---------------------------------------------------------------------------
// GEMM2: H2 = relu(agg1 + b1) @ W2   (100000x128 @ 128x16), fp32 WMMA 16x16x4
// block = 128 threads = 4 waves; each wave owns one 16-node row tile and a
// private LDS region (padded stride 132). Bias+ReLU fused into the staging.
// No cross-wave LDS sharing -> s_wait_dscnt 0 suffices (LDS in-order per wave).
// ---------------------------------------------------------------------------
__global__ __launch_bounds__(128) void k_gemm2(const float* __restrict__ A,
                                               const float* __restrict__ b1,
                                               const float* __restrict__ W2,
                                               float* __restrict__ H2) {
    __shared__ float lds2[4][16 * 132];
    const int t    = threadIdx.x;
    const int lane = t & 31;
    const int wave = t >> 5;
    const int mt   = blockIdx.x * 4 + wave;    // row-tile index (wave-uniform)
    if (mt >= MTILES) return;                  // whole-wave exit, EXEC stays full
    const int Mb = mt * 16;

    // stage 16x128 tile with fused bias + ReLU (coalesced)
#pragma unroll
    for (int i = 0; i < 64; ++i) {
        int idx = lane + 32 * i;
        int row = idx >> 7;
        int col = idx & 127;
        float v = A[(size_t)(Mb + row) * F_HID + col] + b1[col];
        lds2[wave][row * 132 + col] = (v > 0.0f) ? v : 0.0f;
    }
    asm volatile("s_wait_dscnt 0" ::: "memory");

    const int r    = lane & 15;
    const int koff = (lane >> 4) << 1;
    v8f acc = {};
#pragma unroll 8
    for (int k0 = 0; k0 < F_HID; k0 += 4) {
        v2f a, b;
        a.x = lds2[wave][r * 132 + k0 + koff];
        a.y = lds2[wave][r * 132 + k0 + koff + 1];
        b.x = W2[(k0 + koff)     * F_OUT + r];
        b.y = W2[(k0 + koff + 1) * F_OUT + r];
        acc = __builtin_amdgcn_wmma_f32_16x16x4_f32(
                  false, a, false, b, (short)0, acc, false, false);
    }

    const int col   = r;
    const int rbase = Mb + ((lane >> 4) << 3);
#pragma unroll
    for (int j = 0; j < 8; ++j)
        H2[(size_t)(rbase + j) * F_OUT + col] = acc[j];
}

// out[n][c] = b2[c]  (then scatter2 accumulates on top -> agg + b2)
__global__ void k_init_out(const float* __restrict__ b2, float* __restrict__ out) {
    int i = blockIdx.x * 256 + threadIdx.x;
    if (i < N_NODES * F_OUT) out[i] = b2[i & 15];
}

__global__ void k_scatter2(const long long* __restrict__ src,
                           const long long* __restrict__ dst,
                           const float* __restrict__ dinv,
                           const float* __restrict__ H2,
                           float* __restrict__ out) {
    long long gid = (long long)blockIdx.x * 256 + threadIdx.x;
    int e = (int)(gid >> 4);
    int c = (int)(gid & 15);
    if (e >= N_EDGES + N_NODES) return;
    int s, d;
    if (e < N_EDGES) { s = (int)src[e]; d = (int)dst[e]; }
    else             { s = d = e - N_EDGES; }
    float w = dinv[s] * dinv[d];
    float v = H2[(size_t)s * F_OUT + c] * w;
    __hip_atomic_fetch_add(&out[(size_t)d * F_OUT + c], v,
                           __ATOMIC_RELAXED, __HIP_MEMORY_SCOPE_AGENT);
}

// ---------------------------------------------------------------------------
// launch: deg -> dinv -> zero agg1 -> GEMM1 -> scatter1 -> GEMM2
//         -> out=b2 -> scatter2
// workspace usage ~110 MB:
//   [0)            deg/dinv  (100000 f32, slot rounded to 512 KB)
//   [512KB)        H1        (100000x128 f32 = 51.2 MB)
//   [+51.2MB)      agg1      (51.2 MB)
//   [+51.2MB)      H2        (100000x16 f32 = 6.4 MB)
// ---------------------------------------------------------------------------
extern "C" void kernel_launch(void* const* d_in, const int* in_sizes, int n_in,
                              void* d_out, int out_size, void* d_ws, size_t ws_size,
                              hipStream_t stream) {
    const float*     X  = (const float*)d_in[0];
    const long long* ei = (const long long*)d_in[1];   // int64 edge_index [2, E]
    const float*     W1 = (const float*)d_in[2];
    const float*     b1 = (const float*)d_in[3];
    const float*     W2 = (const float*)d_in[4];
    const float*     b2 = (const float*)d_in[5];
    float*           out = (float*)d_out;

    const long long* src = ei;            // edge_index[0]
    const long long* dst = ei + N_EDGES;  // edge_index[1]

    float* deg  = (float*)d_ws;                         // also holds dinv (in place)
    float* H1   = deg + (512 * 1024 / 4);
    float* agg1 = H1 + (size_t)N_NODES * F_HID;
    float* H2   = agg1 + (size_t)N_NODES * F_HID;

    // normalization coefficients
    k_init_deg <<<(N_NODES + 255) / 256, 256, 0, stream>>>(deg);
    k_count_deg<<<(N_EDGES + 255) / 256, 256, 0, stream>>>(dst, deg);
    k_dinv     <<<(N_NODES + 255) / 256, 256, 0, stream>>>(deg);

    // layer 1
    int n4 = N_NODES * F_HID / 4;
    k_zero4    <<<(n4 + 255) / 256, 256, 0, stream>>>((float4*)agg1, n4);
    k_gemm1    <<<MTILES, 256, 0, stream>>>(X, W1, H1);
    long long T1 = (long long)(N_EDGES + N_NODES) * F_HID;
    k_scatter1 <<<(unsigned)((T1 + 255) / 256), 256, 0, stream>>>(src, dst, deg, H1, agg1);

    // layer 2 (bias + ReLU fused into GEMM2 staging)
    k_gemm2    <<<(MTILES + 3) / 4, 128, 0, stream>>>(agg1, b1, W2, H2);
    k_init_out <<<(N_NODES * F_OUT + 255) / 256, 256, 0, stream>>>(b2, out);
    long long T2 = (long long)(N_EDGES + N_NODES) * F_OUT;
    k_scatter2 <<<(unsigned)((T2 + 255) / 256), 256, 0, stream>>>(src, dst, deg, H2, out);
}